// Head_42202348650528
// MI455X (gfx1250) — compile-verified
//
#include <hip/hip_runtime.h>

// MI455X / gfx1250. Workload is tiny (~29 MFLOP, ~5.5 MB traffic -> ~240 ns
// at 23.3 TB/s HBM): purely bandwidth/launch-bound, so keep full fp32 and use
// V_WMMA_F32_16X16X4_F32 chains (K=4 x 4 = K=16) for every 16x16x16 GEMM tile.

typedef float v2f __attribute__((ext_vector_type(2)));
typedef float v8f __attribute__((ext_vector_type(8)));

#define NPATCH   16
#define NE       16
#define TTOT     16
#define R_TOTAL  28528          // sum_{T=1..16} (16*(T+1)^2 - 1)
#define KV_TILES 1784           // sum_{T=1..16} (T+1)^2
#define Q_TILES  16

// ---------------------------------------------------------------------------
// Builder: replay the Python index loops. One thread per (T, j) segment.
// Segment base is closed-form: table base B_T = sum_{T'<T} (16*(T'+1)^2 - 1),
// within-table base_j = j*(T+1)^2 - (j>0). Entries packed as time*16 + patch.
// ---------------------------------------------------------------------------
__global__ void build_indices(int* __restrict__ idx_tab) {
    int tid = blockIdx.x * blockDim.x + threadIdx.x;
    if (tid >= 256) return;
    int T = (tid >> 4) + 1;   // prefix length 1..16
    int j = tid & 15;

    int base = 0;
    for (int tp = 1; tp < T; ++tp) base += 16 * (tp + 1) * (tp + 1) - 1;
    int Q   = (T + 1) * (T + 1);
    int pos = base + (j == 0 ? 0 : j * Q - 1);

    idx_tab[pos++] = (T - 1) * 16 + j;          // initial (T-1, j) entry
    int i = 2;
    for (int t = T - 1; t >= 0; --t) {
        for (int k = -i + 1; k <= i - 1; ++k) {
            for (int l = -i + 1; l <= i - 1; ++l) {
                int idx = j + 16 * k + l;
                if (!(j == 0 && l == 0 && i == 2) && idx >= 0 && idx < 16) {
                    idx_tab[pos++] = t * 16 + idx;
                }
            }
        }
        ++i;
    }
}

// ---------------------------------------------------------------------------
// 16x16x16 f32 GEMM via 4 chained V_WMMA_F32_16X16X4_F32.
// B layout per ISA: lane L holds B[K = 4*kk + 2*(L>>4) + v][n = L&15]
//   = W[n][4*kk + 2*hi + v]  -> float2 at wrow + 4*kk (wrow = W + n*16 + 2*hi)
// ---------------------------------------------------------------------------
__device__ __forceinline__ v8f gemm16(const v2f a0, const v2f a1,
                                      const v2f a2, const v2f a3,
                                      const float* __restrict__ wrow, v8f c) {
    v2f b0 = *(const v2f*)(wrow + 0);
    v2f b1 = *(const v2f*)(wrow + 4);
    v2f b2 = *(const v2f*)(wrow + 8);
    v2f b3 = *(const v2f*)(wrow + 12);
    c = __builtin_amdgcn_wmma_f32_16x16x4_f32(false, a0, false, b0, (short)0, c, false, false);
    c = __builtin_amdgcn_wmma_f32_16x16x4_f32(false, a1, false, b1, (short)0, c, false, false);
    c = __builtin_amdgcn_wmma_f32_16x16x4_f32(false, a2, false, b2, (short)0, c, false, false);
    c = __builtin_amdgcn_wmma_f32_16x16x4_f32(false, a3, false, b3, (short)0, c, false, false);
    return c;
}

// One wave32 per 16-row output tile. Blocks [0,16): q tiles; rest: k/v tiles.
__global__ void __launch_bounds__(32)
head_gemm(const float* __restrict__ x,   // (1,16,16,16)
          const float* __restrict__ Wq,  // (16,16)
          const float* __restrict__ Wk,  // (16,16,16)
          const float* __restrict__ bk,  // (16,16)
          const float* __restrict__ Wv,  // (16,16,16)
          const float* __restrict__ bv,  // (16,16)
          const int*   __restrict__ idx_tab,
          float*       __restrict__ out) {
    const int lane = threadIdx.x;
    const int n    = lane & 15;   // N column (B/C/D); also A row M for gather
    const int hi   = lane >> 4;
    const int bid  = blockIdx.x;

    int  src;                 // packed time*16+patch of A row M = n
    bool mask15 = false;      // last tile of a table: row 15 invalid

    if (bid < Q_TILES) {
        src = bid * 16 + n;   // time = bid, patch = n
    } else {
        int rem = bid - Q_TILES;
        int T = 1;
        for (;;) { int q = (T + 1) * (T + 1); if (rem < q) break; rem -= q; ++T; }
        int rowBaseT = 0;
        for (int tp = 1; tp < T; ++tp) rowBaseT += 16 * (tp + 1) * (tp + 1) - 1;
        int Q       = (T + 1) * (T + 1);
        int tileRow = rowBaseT + 16 * rem;            // k-row index of tile row 0
        mask15      = (rem == Q - 1);
        src = idx_tab[tileRow + ((mask15 && n == 15) ? 0 : n)];

        // A gather: patchified x, element e of patch vec lives at
        // time*256 + (pat>>2)*64 + (pat&3)*4 + (e>>2)*16 + (e&3).
        // Lane needs e = 4*kk + 2*hi + {0,1}: float2 at abase + 16*kk.
        int time = src >> 4, pat = src & 15;
        const float* abase = x + time * 256 + (pat >> 2) * 64 + (pat & 3) * 4 + 2 * hi;
        v2f a0 = *(const v2f*)(abase + 0);
        v2f a1 = *(const v2f*)(abase + 16);
        v2f a2 = *(const v2f*)(abase + 32);
        v2f a3 = *(const v2f*)(abase + 48);

        int t = T - 1;
        // --- K ---
        float bkv = bk[t * 16 + n];
        v8f ck = {bkv, bkv, bkv, bkv, bkv, bkv, bkv, bkv};
        ck = gemm16(a0, a1, a2, a3, Wk + t * 256 + n * 16 + 2 * hi, ck);
        // --- V ---
        float bvv = bv[t * 16 + n];
        v8f cv = {bvv, bvv, bvv, bvv, bvv, bvv, bvv, bvv};
        cv = gemm16(a0, a1, a2, a3, Wv + t * 256 + n * 16 + 2 * hi, cv);

        // D layout: lane L, vgpr g -> D[M = g + 8*hi][N = L&15]
        float* okrow = out + (256 + tileRow) * 16;
        float* ovrow = out + (256 + R_TOTAL + tileRow) * 16;
#pragma unroll
        for (int g = 0; g < 8; ++g) {
            int m = g + 8 * hi;
            if (!(mask15 && m == 15)) {
                okrow[m * 16 + n] = ck[g];
                ovrow[m * 16 + n] = cv[g];
            }
        }
        return;
    }

    // ---- q tile ----
    int time = src >> 4, pat = src & 15;
    const float* abase = x + time * 256 + (pat >> 2) * 64 + (pat & 3) * 4 + 2 * hi;
    v2f a0 = *(const v2f*)(abase + 0);
    v2f a1 = *(const v2f*)(abase + 16);
    v2f a2 = *(const v2f*)(abase + 32);
    v2f a3 = *(const v2f*)(abase + 48);

    v8f c = {0.f, 0.f, 0.f, 0.f, 0.f, 0.f, 0.f, 0.f};
    c = gemm16(a0, a1, a2, a3, Wq + n * 16 + 2 * hi, c);

    float* orow = out + (bid * 16) * 16;
#pragma unroll
    for (int g = 0; g < 8; ++g) {
        int m = g + 8 * hi;
        orow[m * 16 + n] = c[g];
    }
}

extern "C" void kernel_launch(void* const* d_in, const int* in_sizes, int n_in,
                              void* d_out, int out_size, void* d_ws, size_t ws_size,
                              hipStream_t stream) {
    const float* x  = (const float*)d_in[0];
    const float* Wq = (const float*)d_in[1];
    const float* Wk = (const float*)d_in[2];
    const float* bk = (const float*)d_in[3];
    const float* Wv = (const float*)d_in[4];
    const float* bv = (const float*)d_in[5];

    int* idx_tab = (int*)d_ws;   // needs R_TOTAL * 4 = ~112 KB of scratch

    build_indices<<<2, 128, 0, stream>>>(idx_tab);
    head_gemm<<<Q_TILES + KV_TILES, 32, 0, stream>>>(
        x, Wq, Wk, bk, Wv, bv, idx_tab, (float*)d_out);
}